// Model_36223754174929
// MI455X (gfx1250) — compile-verified
//
#include <hip/hip_runtime.h>
#include <math.h>

typedef __attribute__((ext_vector_type(2))) float v2f;
typedef __attribute__((ext_vector_type(8))) float v8f;

#define T_STEPS 512
#define BATCH   512
#define EMB     32
#define HID     32
#define FF      16
#define NCLS    7
#define ROWS    16   // batch rows per block (WMMA M)

__device__ __forceinline__ float sigm(float v)  { return 1.0f / (1.0f + __expf(-v)); }
__device__ __forceinline__ float tanhx(float v) { return 2.0f / (1.0f + __expf(-2.0f * v)) - 1.0f; }

// One wave (32 threads) per block; block b owns batch rows [16b, 16b+16).
// Runs the full embed -> LSTM -> FF -> logits pipeline for all 512 timesteps,
// then does the axis-0 (time) softmax for its own rows. fp32 WMMA throughout.
__global__ __launch_bounds__(32)
void lstm_fused_kernel(const int* __restrict__ x,
                       const float* __restrict__ emb,
                       const float* __restrict__ W_ih,
                       const float* __restrict__ W_hh,
                       const float* __restrict__ b_ih,
                       const float* __restrict__ b_hh,
                       const float* __restrict__ W1,
                       const float* __restrict__ b1,
                       const float* __restrict__ W2,
                       const float* __restrict__ b2,
                       float* __restrict__ out)
{
    // Weight matrices pre-swizzled into WMMA B-fragment order:
    // fragment f = 64 floats; lane l reads {f*64 + 2l, f*64 + 2l + 1}.
    __shared__ float sWg[16 * 8 * 64];  // gates: K=64 (16 ktiles) x N=128 (8 ntiles)
    __shared__ float sW1[8 * 64];       // FF:    K=32 (8 ktiles)  x N=16
    __shared__ float sW2[4 * 64];       // head:  K=16 (4 ktiles)  x N=16 (cols>=7 zero)
    __shared__ float sH[ROWS * HID];    // h_t tile, row-major 16x32
    __shared__ float sZ[ROWS * FF];     // z tile, row-major 16x16

    const int lane  = threadIdx.x;
    const int half  = lane >> 4;        // 0/1
    const int l16   = lane & 15;
    const int kb    = half * 2;         // K sub-offset inside a K=4 slab
    const int mbase = half * 8;         // D-layout: row = mbase + r
    const int mrow  = l16;              // A-layout: this lane's matrix row
    const int bm0   = blockIdx.x * ROWS;

    // ---- build B fragments in LDS (combined [W_ih^T ; W_hh^T], K=64 x N=128) ----
    for (int kt = 0; kt < 16; ++kt) {
        for (int nt = 0; nt < 8; ++nt) {
            const int n = nt * 16 + l16;
            const int k = kt * 4 + kb;
            float w0, w1;
            if (k < EMB) { w0 = W_ih[n * EMB + k];         w1 = W_ih[n * EMB + k + 1]; }
            else         { w0 = W_hh[n * HID + (k - EMB)]; w1 = W_hh[n * HID + (k - EMB) + 1]; }
            sWg[(kt * 8 + nt) * 64 + lane * 2 + 0] = w0;
            sWg[(kt * 8 + nt) * 64 + lane * 2 + 1] = w1;
        }
    }
    for (int kt = 0; kt < 8; ++kt) {
        const int k = kt * 4 + kb;
        sW1[kt * 64 + lane * 2 + 0] = W1[l16 * HID + k];
        sW1[kt * 64 + lane * 2 + 1] = W1[l16 * HID + k + 1];
    }
    for (int kt = 0; kt < 4; ++kt) {
        const int k = kt * 4 + kb;
        float w0 = 0.0f, w1 = 0.0f;
        if (l16 < NCLS) { w0 = W2[l16 * FF + k]; w1 = W2[l16 * FF + k + 1]; }
        sW2[kt * 64 + lane * 2 + 0] = w0;
        sW2[kt * 64 + lane * 2 + 1] = w1;
    }
    for (int i = lane; i < ROWS * HID; i += 32) sH[i] = 0.0f;   // h0 = 0

    // per-lane bias values in D layout (bias depends on N only)
    float bg[8];
    #pragma unroll
    for (int nt = 0; nt < 8; ++nt) bg[nt] = b_ih[nt * 16 + l16] + b_hh[nt * 16 + l16];
    const float b1v = b1[l16];
    const float b2v = (l16 < NCLS) ? b2[l16] : 0.0f;

    // cell state: c for rows (mbase+r), hidden units j0=l16 and j1=16+l16
    float c0r[8], c1r[8];
    #pragma unroll
    for (int r = 0; r < 8; ++r) { c0r[r] = 0.0f; c1r[r] = 0.0f; }

    __syncthreads();

    for (int t = 0; t < T_STEPS; ++t) {
        // ---- A fragments: A = [e_tile | h_tile], 16 x 64 ----
        v2f afr[16];
        const int idx = x[t * BATCH + bm0 + mrow];
        const float* erow = emb + idx * EMB;
        #pragma unroll
        for (int kt = 0; kt < 8; ++kt) {            // embedding half (global gather)
            afr[kt].x = erow[kt * 4 + kb];
            afr[kt].y = erow[kt * 4 + kb + 1];
        }
        #pragma unroll
        for (int kt = 0; kt < 8; ++kt) {            // h_{t-1} half (LDS)
            afr[8 + kt].x = sH[mrow * HID + kt * 4 + kb];
            afr[8 + kt].y = sH[mrow * HID + kt * 4 + kb + 1];
        }

        // ---- gates = A @ [W_ih^T;W_hh^T] + (b_ih+b_hh) ----
        v8f acc[8];
        #pragma unroll
        for (int nt = 0; nt < 8; ++nt) {
            #pragma unroll
            for (int r = 0; r < 8; ++r) acc[nt][r] = bg[nt];
        }
        #pragma unroll
        for (int kt = 0; kt < 16; ++kt) {
            #pragma unroll
            for (int nt = 0; nt < 8; ++nt) {
                v2f bf;
                bf.x = sWg[(kt * 8 + nt) * 64 + lane * 2 + 0];
                bf.y = sWg[(kt * 8 + nt) * 64 + lane * 2 + 1];
                acc[nt] = __builtin_amdgcn_wmma_f32_16x16x4_f32(
                    false, afr[kt], false, bf, (short)0, acc[nt], false, false);
            }
        }

        // ---- elementwise LSTM update (i,f,g,o = ntiles {0,1},{2,3},{4,5},{6,7}) ----
        float hnew0[8], hnew1[8];
        #pragma unroll
        for (int r = 0; r < 8; ++r) {
            float i0 = sigm(acc[0][r]), f0 = sigm(acc[2][r]);
            float g0 = tanhx(acc[4][r]), o0 = sigm(acc[6][r]);
            float c0 = f0 * c0r[r] + i0 * g0;  c0r[r] = c0;
            hnew0[r] = o0 * tanhx(c0);
            float i1 = sigm(acc[1][r]), f1 = sigm(acc[3][r]);
            float g1 = tanhx(acc[5][r]), o1 = sigm(acc[7][r]);
            float c1 = f1 * c1r[r] + i1 * g1;  c1r[r] = c1;
            hnew1[r] = o1 * tanhx(c1);
        }
        __syncthreads();                 // everyone finished reading old sH
        #pragma unroll
        for (int r = 0; r < 8; ++r) {
            sH[(mbase + r) * HID + l16]      = hnew0[r];
            sH[(mbase + r) * HID + 16 + l16] = hnew1[r];
        }
        __syncthreads();                 // new h_t visible

        // ---- z = relu(h_t @ W1^T + b1), 16x16 ----
        v8f zacc;
        #pragma unroll
        for (int r = 0; r < 8; ++r) zacc[r] = b1v;
        #pragma unroll
        for (int kt = 0; kt < 8; ++kt) {
            v2f af, bf;
            af.x = sH[mrow * HID + kt * 4 + kb];
            af.y = sH[mrow * HID + kt * 4 + kb + 1];
            bf.x = sW1[kt * 64 + lane * 2 + 0];
            bf.y = sW1[kt * 64 + lane * 2 + 1];
            zacc = __builtin_amdgcn_wmma_f32_16x16x4_f32(
                false, af, false, bf, (short)0, zacc, false, false);
        }
        #pragma unroll
        for (int r = 0; r < 8; ++r)
            sZ[(mbase + r) * FF + l16] = fmaxf(zacc[r], 0.0f);
        __syncthreads();

        // ---- logits = z @ W2^T + b2 (16x16, cols >= 7 are garbage/unused) ----
        v8f lacc;
        #pragma unroll
        for (int r = 0; r < 8; ++r) lacc[r] = b2v;
        #pragma unroll
        for (int kt = 0; kt < 4; ++kt) {
            v2f af, bf;
            af.x = sZ[mrow * FF + kt * 4 + kb];
            af.y = sZ[mrow * FF + kt * 4 + kb + 1];
            bf.x = sW2[kt * 64 + lane * 2 + 0];
            bf.y = sW2[kt * 64 + lane * 2 + 1];
            lacc = __builtin_amdgcn_wmma_f32_16x16x4_f32(
                false, af, false, bf, (short)0, lacc, false, false);
        }
        if (l16 < NCLS) {
            #pragma unroll
            for (int r = 0; r < 8; ++r) {
                const int b = bm0 + mbase + r;
                out[t * (BATCH * NCLS) + b * NCLS + l16] = lacc[r];
            }
        }
        __syncthreads();                 // sZ/sH reuse next iteration
    }

    // make our own logit stores visible before re-reading them
    asm volatile("s_wait_storecnt 0" ::: "memory");
    __syncthreads();

    // ---- softmax over the TIME axis for the 16*7 columns this block owns ----
    for (int q = lane; q < ROWS * NCLS; q += 32) {
        const int b = bm0 + q / NCLS;
        const int c = q % NCLS;
        float* p = out + b * NCLS + c;
        const int stride = BATCH * NCLS;
        float mx = -INFINITY;
        for (int t = 0; t < T_STEPS; ++t) mx = fmaxf(mx, p[t * stride]);
        float s = 0.0f;
        for (int t = 0; t < T_STEPS; ++t) s += __expf(p[t * stride] - mx);
        const float inv = 1.0f / s;
        for (int t = 0; t < T_STEPS; ++t) p[t * stride] = __expf(p[t * stride] - mx) * inv;
    }
}

extern "C" void kernel_launch(void* const* d_in, const int* in_sizes, int n_in,
                              void* d_out, int out_size, void* d_ws, size_t ws_size,
                              hipStream_t stream) {
    (void)in_sizes; (void)n_in; (void)d_ws; (void)ws_size; (void)out_size;
    const int*   x    = (const int*)  d_in[0];
    const float* emb  = (const float*)d_in[1];
    const float* W_ih = (const float*)d_in[2];
    const float* W_hh = (const float*)d_in[3];
    const float* b_ih = (const float*)d_in[4];
    const float* b_hh = (const float*)d_in[5];
    const float* W1   = (const float*)d_in[6];
    const float* b1   = (const float*)d_in[7];
    const float* W2   = (const float*)d_in[8];
    const float* b2   = (const float*)d_in[9];
    float* out = (float*)d_out;

    dim3 grid(BATCH / ROWS);   // 32 blocks, one wave each (16 batch rows/block)
    dim3 block(32);
    lstm_fused_kernel<<<grid, block, 0, stream>>>(x, emb, W_ih, W_hh, b_ih, b_hh,
                                                  W1, b1, W2, b2, out);
}